// GNNEncoder_8693013807597
// MI455X (gfx1250) — compile-verified
//
#include <hip/hip_runtime.h>

#define FDIM 128   // EMB == HID == OUT == 128

typedef __attribute__((ext_vector_type(2))) float v2f;
typedef __attribute__((ext_vector_type(8))) float v8f;

// Non-returning hardware f32 atomic add (global_atomic_add_f32, STOREcnt).
__device__ __forceinline__ void hw_atomic_add_f32(float* p, float v) {
    asm volatile("global_atomic_add_f32 %0, %1, off" : : "v"(p), "v"(v) : "memory");
}

// ---------------------------------------------------------------------------
// Degree / normalization
// ---------------------------------------------------------------------------
__global__ __launch_bounds__(256) void k_deg_init(float* __restrict__ deg, int N) {
    int i = blockIdx.x * 256 + threadIdx.x;
    if (i < N) deg[i] = 1.0f;  // self-loop weight
}

__global__ __launch_bounds__(256) void k_deg_edges(const int* __restrict__ dst,
                                                   const float* __restrict__ w,
                                                   float* __restrict__ deg, int E) {
    int e = blockIdx.x * 256 + threadIdx.x;
    if (e < E) hw_atomic_add_f32(&deg[dst[e]], w[e]);
}

__global__ __launch_bounds__(256) void k_dinv(const float* __restrict__ deg,
                                              float* __restrict__ dinv, int N) {
    int i = blockIdx.x * 256 + threadIdx.x;
    if (i < N) {
        float d = deg[i];
        dinv[i] = d > 0.0f ? rsqrtf(d) : 0.0f;
    }
}

// ---------------------------------------------------------------------------
// Tiled GEMM: H[16-row tile] = X_tile @ W  via V_WMMA_F32_16X16X4_F32.
// Tile staged into LDS with GLOBAL_LOAD_ASYNC_TO_LDS_B128 (ASYNCcnt path).
// MODE 0: X row r gathered from emb_table[labels[r]]  (layer-1 input)
// MODE 1: X row r = relu(Xsrc[r]), relu applied at fragment read
// Block = 256 threads = 8 waves; wave w computes output cols [16w, 16w+16).
// ---------------------------------------------------------------------------
template <int MODE>
__global__ __launch_bounds__(256) void k_gemm(const float* __restrict__ Xsrc,
                                              const int* __restrict__ labels,
                                              const float* __restrict__ emb,
                                              const float* __restrict__ W,
                                              float* __restrict__ H, int N) {
    __shared__ float Xs[16][FDIM + 4];

    const int tid  = threadIdx.x;
    const int row0 = blockIdx.x * 16;

    // Async-stage 16x128 f32 tile: each thread moves 8 floats (2x b128),
    // memory -> LDS directly, no VGPR round trip.
    {
        int r   = tid >> 4;          // 0..15
        int c0  = (tid & 15) * 8;    // 0,8,...,120
        int row = row0 + r;
        if (row > N - 1) row = N - 1;  // tail clamp (N=50000 is multiple of 16)
        const float* gsrc;
        if (MODE == 0) gsrc = emb + (size_t)labels[row] * FDIM + c0;
        else           gsrc = Xsrc + (size_t)row * FDIM + c0;
        // LDS aperture address: low 32 bits of generic pointer (ISA 10.2).
        unsigned l0 = (unsigned)(uintptr_t)&Xs[r][c0];
        asm volatile("global_load_async_to_lds_b128 %0, %1, off"
                     : : "v"(l0), "v"(gsrc) : "memory");
        asm volatile("global_load_async_to_lds_b128 %0, %1, off offset:16"
                     : : "v"(l0), "v"(gsrc) : "memory");
        asm volatile("s_wait_asynccnt 0x0" : : : "memory");
    }
    __syncthreads();

    const int wave = tid >> 5;
    const int lane = tid & 31;
    const int n0   = wave * 16;

    // ISA f32 A 16x4 layout: lanes 0-15 -> M=lane, K = {k0, k0+1};
    //                        lanes 16-31 -> M=lane-16, K = {k0+2, k0+3}.
    const int ml   = lane & 15;
    const int koff = (lane >> 4) << 1;

    v8f acc = {};
#pragma unroll
    for (int k0 = 0; k0 < FDIM; k0 += 4) {
        v2f a, b;
        float a0 = Xs[ml][k0 + koff];
        float a1 = Xs[ml][k0 + koff + 1];
        if (MODE == 1) {               // fused ReLU on layer-1 activations
            a0 = fmaxf(a0, 0.0f);
            a1 = fmaxf(a1, 0.0f);
        }
        a.x = a0;
        a.y = a1;
        const float* wp = W + (k0 + koff) * FDIM + n0 + ml;  // B 4x16: row striped over lanes
        b.x = wp[0];
        b.y = wp[FDIM];
        acc = __builtin_amdgcn_wmma_f32_16x16x4_f32(
            /*neg_a=*/false, a, /*neg_b=*/false, b,
            /*c_mod=*/(short)0, acc, /*reuse_a=*/false, /*reuse_b=*/false);
    }

    // C/D 16x16 f32 layout: VGPR j -> (M=j, N=lane) lanes 0-15,
    //                                 (M=j+8, N=lane-16) lanes 16-31.
#pragma unroll
    for (int j = 0; j < 8; ++j) {
        int m = row0 + j + ((lane >> 4) << 3);
        if (m < N) H[(size_t)m * FDIM + n0 + ml] = acc[j];
    }
}

// ---------------------------------------------------------------------------
// agg[i,c] = bias[c] + H[i,c] * dinv[i]^2   (bias + self-loop message)
// ---------------------------------------------------------------------------
__global__ __launch_bounds__(256) void k_agg_init(const float* __restrict__ H,
                                                  const float* __restrict__ dinv,
                                                  const float* __restrict__ bias,
                                                  float* __restrict__ agg, int N) {
    int idx = blockIdx.x * 256 + threadIdx.x;
    if (idx < N * FDIM) {
        int i = idx >> 7;
        int c = idx & (FDIM - 1);
        float di = dinv[i];
        agg[idx] = bias[c] + H[idx] * di * di;
    }
}

// ---------------------------------------------------------------------------
// Edge scatter: one wave per edge; lane handles 4 contiguous features.
// agg[dst] += H[src] * (dinv[src] * w * dinv[dst])  via global_atomic_add_f32
// ---------------------------------------------------------------------------
__global__ __launch_bounds__(256) void k_scatter(const int* __restrict__ src,
                                                 const int* __restrict__ dst,
                                                 const float* __restrict__ w,
                                                 const float* __restrict__ dinv,
                                                 const float* __restrict__ H,
                                                 float* __restrict__ agg, int E) {
    int e    = (blockIdx.x * 256 + threadIdx.x) >> 5;
    int lane = threadIdx.x & 31;
    if (e >= E) return;

    int   s     = src[e];
    int   d     = dst[e];
    float coeff = dinv[s] * w[e] * dinv[d];

    const float4 h = *reinterpret_cast<const float4*>(H + (size_t)s * FDIM + lane * 4);
    float* out = agg + (size_t)d * FDIM + lane * 4;
    hw_atomic_add_f32(out + 0, h.x * coeff);
    hw_atomic_add_f32(out + 1, h.y * coeff);
    hw_atomic_add_f32(out + 2, h.z * coeff);
    hw_atomic_add_f32(out + 3, h.w * coeff);
}

// ---------------------------------------------------------------------------
extern "C" void kernel_launch(void* const* d_in, const int* in_sizes, int n_in,
                              void* d_out, int out_size, void* d_ws, size_t ws_size,
                              hipStream_t stream) {
    const int*   labels = (const int*)d_in[0];
    const int*   eidx   = (const int*)d_in[1];
    const float* weight = (const float*)d_in[2];
    const float* emb    = (const float*)d_in[3];
    const float* W1     = (const float*)d_in[4];
    const float* b1     = (const float*)d_in[5];
    const float* W2     = (const float*)d_in[6];
    const float* b2     = (const float*)d_in[7];

    const int N = in_sizes[0];
    const int E = in_sizes[2];
    const int* src = eidx;      // edge_index[0]
    const int* dst = eidx + E;  // edge_index[1]

    // Workspace layout (floats): deg[N] | dinv[N] | H[N*128] | agg1[N*128]
    float* ws   = (float*)d_ws;
    float* deg  = ws;
    float* dinv = deg + N;
    float* H    = dinv + N;
    float* agg1 = H + (size_t)N * FDIM;
    float* out  = (float*)d_out;

    const int nbN   = (N + 255) / 256;
    const int nbE   = (E + 255) / 256;
    const int nbNF  = (N * FDIM + 255) / 256;
    const int nbEW  = (E + 7) / 8;        // 8 edges (waves) per block
    const int tiles = (N + 15) / 16;

    // Normalization
    k_deg_init <<<nbN, 256, 0, stream>>>(deg, N);
    k_deg_edges<<<nbE, 256, 0, stream>>>(dst, weight, deg, E);
    k_dinv     <<<nbN, 256, 0, stream>>>(deg, dinv, N);

    // Layer 1: H = emb[labels] @ W1 ; agg1 = b1 + self-loop + edge messages
    k_gemm<0>  <<<tiles, 256, 0, stream>>>(nullptr, labels, emb, W1, H, N);
    k_agg_init <<<nbNF, 256, 0, stream>>>(H, dinv, b1, agg1, N);
    k_scatter  <<<nbEW, 256, 0, stream>>>(src, dst, weight, dinv, H, agg1, E);

    // Layer 2: H = relu(agg1) @ W2 ; out = b2 + self-loop + edge messages
    k_gemm<1>  <<<tiles, 256, 0, stream>>>(agg1, nullptr, nullptr, W2, H, N);
    k_agg_init <<<nbNF, 256, 0, stream>>>(H, dinv, b2, out, N);
    k_scatter  <<<nbEW, 256, 0, stream>>>(src, dst, weight, dinv, H, out, E);
}